// VectLayer_73813307949742
// MI455X (gfx1250) — compile-verified
//
#include <hip/hip_runtime.h>

#define KDE_BATCH   64
#define KDE_NPTS    2048
#define KDE_GRID    2500
#define KDE_GPAIRS  79       // ceil(ceil(2500/16)/2): 32 grid columns per wave

typedef float v2f __attribute__((ext_vector_type(2)));
typedef float v8f __attribute__((ext_vector_type(8)));

// One wave (32 threads) per (batch b, 32-wide grid-column pair of tiles).
// WMMA f32 16x16x4 computes a full log2-space exponent tile per grid tile:
//   D[n,g] = (25*log2e*px)*gx + (25*log2e*py)*gy + base_n*1 + 1*gg_g
// then kern = exp2(D), out[g] += w_n * kern. The A operand (point data) and
// the LDS weight loads are amortized over two WMMAs (two grid tiles).
__global__ __launch_bounds__(32) void kde_wmma2_kernel(
    const float* __restrict__ sms,   // [B, N, 3] = (px, py, w)
    const float* __restrict__ grid,  // [G, 2]
    float* __restrict__ out)         // [B, G]
{
    constexpr float L2E = 1.4426950408889634f;
    constexpr float CB  = -12.5f * L2E;   // applied to |p|^2 and |g|^2
    constexpr float CS  =  25.0f * L2E;   // applied to the cross term

    __shared__ float wlds[KDE_NPTS];      // 8 KB: weights for this batch

    const int lane = threadIdx.x;         // 0..31 (wave32)
    const int wid  = blockIdx.x;
    const int b    = wid / KDE_GPAIRS;
    const int gp   = wid - b * KDE_GPAIRS;
    const int g0   = gp * 32;

    const float* __restrict__ smsb = sms + (size_t)b * (KDE_NPTS * 3);

    // Stage weights (sms[...,2]) into LDS once per wave.
    for (int i = lane; i < KDE_NPTS; i += 32)
        wlds[i] = smsb[3 * i + 2];
    __syncthreads();

    const bool hi = (lane >= 16);
    const int  lo = lane & 15;

    // Two loop-invariant B operands (grid tiles 0 and 1 of this pair).
    // B (4x16) half-lane layout: VGPR0 = {K0 | K2}, VGPR1 = {K1 | K3}.
    // Padded columns (g >= 2500) -> gg = -1e30 => exp2 -> 0.
    v2f B0, B1;
    {
        int   gA = g0 + lo;
        float gxA = 0.0f, gyA = 0.0f, ggA = -1.0e30f;
        if (gA < KDE_GRID) {
            gxA = grid[2 * gA];
            gyA = grid[2 * gA + 1];
            ggA = CB * fmaf(gyA, gyA, gxA * gxA);
        }
        B0.x = hi ? 1.0f : gxA;   // K0 = gx, K2 = 1
        B0.y = hi ? ggA  : gyA;   // K1 = gy, K3 = gg

        int   gB = g0 + 16 + lo;
        float gxB = 0.0f, gyB = 0.0f, ggB = -1.0e30f;
        if (gB < KDE_GRID) {
            gxB = grid[2 * gB];
            gyB = grid[2 * gB + 1];
            ggB = CB * fmaf(gyB, gyB, gxB * gxB);
        }
        B1.x = hi ? 1.0f : gxB;
        B1.y = hi ? ggB  : gyB;
    }

    float acc0[8], acc1[8];
#pragma unroll
    for (int r = 0; r < 8; ++r) { acc0[r] = 0.0f; acc1[r] = 0.0f; }

    const int wofs = hi ? 8 : 0;  // D rows 0-7 in lanes 0-15, rows 8-15 in lanes 16-31

    for (int n0 = 0; n0 < KDE_NPTS; n0 += 16) {
        const float* __restrict__ p = smsb + 3 * (n0 + lo);
        // prefetch next n-tile's point data (speculative, OOB-safe)
        __builtin_prefetch(p + 48, 0, 1);

        float px   = p[0];
        float py   = p[1];
        float base = CB * fmaf(py, py, px * px);

        // A (16x4) half-lane layout: VGPR0 = {K0 | K2}, VGPR1 = {K1 | K3}
        v2f A;
        A.x = hi ? base : CS * px;  // K0 = px', K2 = base
        A.y = hi ? 1.0f : CS * py;  // K1 = py', K3 = 1

        v8f Z = {0.f, 0.f, 0.f, 0.f, 0.f, 0.f, 0.f, 0.f};
        v8f D0 = __builtin_amdgcn_wmma_f32_16x16x4_f32(
                     false, A, false, B0, (short)0, Z, false, false);
        v8f D1 = __builtin_amdgcn_wmma_f32_16x16x4_f32(
                     false, A, false, B1, (short)0, Z, false, false);

        // Weights for this half's 8 rows: two broadcast ds_load_b128,
        // shared by both grid tiles.
        const float4 w0 = *(const float4*)(&wlds[n0 + wofs]);
        const float4 w1 = *(const float4*)(&wlds[n0 + wofs + 4]);
        float w[8] = { w0.x, w0.y, w0.z, w0.w, w1.x, w1.y, w1.z, w1.w };

#pragma unroll
        for (int r = 0; r < 8; ++r)
            acc0[r] = fmaf(w[r], __builtin_amdgcn_exp2f(D0[r]), acc0[r]);
#pragma unroll
        for (int r = 0; r < 8; ++r)
            acc1[r] = fmaf(w[r], __builtin_amdgcn_exp2f(D1[r]), acc1[r]);
    }

    float s0 = ((acc0[0] + acc0[1]) + (acc0[2] + acc0[3])) +
               ((acc0[4] + acc0[5]) + (acc0[6] + acc0[7]));
    float s1 = ((acc1[0] + acc1[1]) + (acc1[2] + acc1[3])) +
               ((acc1[4] + acc1[5]) + (acc1[6] + acc1[7]));
    // combine rows 0-7 (lanes 0-15) with rows 8-15 (lanes 16-31)
    s0 += __shfl_xor(s0, 16, 32);
    s1 += __shfl_xor(s1, 16, 32);

    // lanes 0-15 hold tile0 column sums, lanes 16-31 hold tile1's:
    // one fully coalesced 32-lane store over 32 consecutive grid columns.
    float sv = hi ? s1 : s0;
    int   g  = g0 + lane;
    if (g < KDE_GRID)
        out[(size_t)b * KDE_GRID + g] = sv;
}

extern "C" void kernel_launch(void* const* d_in, const int* in_sizes, int n_in,
                              void* d_out, int out_size, void* d_ws, size_t ws_size,
                              hipStream_t stream) {
    (void)in_sizes; (void)n_in; (void)d_ws; (void)ws_size; (void)out_size;
    const float* sms  = (const float*)d_in[0];   // [64, 2048, 3] f32
    const float* grid = (const float*)d_in[1];   // [2500, 2] f32
    float* out = (float*)d_out;                  // [64, 2500] f32

    dim3 gsz(KDE_BATCH * KDE_GPAIRS);            // 5056 waves, one per block
    kde_wmma2_kernel<<<gsz, 32, 0, stream>>>(sms, grid, out);
}